// GaussianSplattingRenderer_72911364817586
// MI455X (gfx1250) — compile-verified
//
#include <hip/hip_runtime.h>
#include <hip/hip_bf16.h>

#define IMG 256
#define NG  1024
#define TTHR  1.0e-4f
#define ACLIP 0.99f

typedef __attribute__((ext_vector_type(2))) float v2f;
typedef __attribute__((ext_vector_type(4))) float v4f;
typedef __attribute__((ext_vector_type(8))) float v8f;
typedef __attribute__((ext_vector_type(4))) int   v4i;

// ---------------------------------------------------------------------------
// Kernel 1: bitonic sort by depth (stable via index tiebreak) + precompute
// inverse-covariance terms for the sorted order. Output layout in d_ws:
//   v4f geom[1024]  : {mx, my, inv00, invoff}        (16 KB)
//   v4f colal[1024] : {alpha, r, g, b}               (16 KB)
//   f32 inv11[1024]                                  ( 4 KB)
// ---------------------------------------------------------------------------
__global__ __launch_bounds__(1024) void gs_sort_prep(
    const float* __restrict__ mean, const float* __restrict__ cov,
    const float* __restrict__ color, const float* __restrict__ alpha,
    const float* __restrict__ depth, v4f* __restrict__ ws4) {
  __shared__ float sd[NG];
  __shared__ int   si[NG];
  const int tid = threadIdx.x;
  sd[tid] = depth[tid];
  si[tid] = tid;
  __syncthreads();
  for (int k = 2; k <= NG; k <<= 1) {
    for (int j = k >> 1; j > 0; j >>= 1) {
      int ixj = tid ^ j;
      if (ixj > tid) {
        float d0 = sd[tid], d1 = sd[ixj];
        int   i0 = si[tid], i1 = si[ixj];
        bool up = ((tid & k) == 0);
        bool gt = (d0 > d1) || (d0 == d1 && i0 > i1);
        if (gt == up) {
          sd[tid] = d1; sd[ixj] = d0;
          si[tid] = i1; si[ixj] = i0;
        }
      }
      __syncthreads();
    }
  }
  const int g = si[tid];
  float a = cov[4 * g + 0], b = cov[4 * g + 1];
  float c = cov[4 * g + 2], d = cov[4 * g + 3];
  float det = a * d - b * c;
  float inv00  = d / det;
  float inv11  = a / det;
  float invoff = -(b + c) / det;

  v4f gm; gm.x = mean[2 * g];  gm.y = mean[2 * g + 1];
  gm.z = inv00;  gm.w = invoff;
  ws4[tid] = gm;

  v4f ca; ca.x = alpha[g];
  ca.y = color[3 * g]; ca.z = color[3 * g + 1]; ca.w = color[3 * g + 2];
  ws4[NG + tid] = ca;

  ((float*)(ws4 + 2 * NG))[tid] = inv11;
}

// ---------------------------------------------------------------------------
// Kernel 2: compositing. One wave = 16 pixels of one row (2 lanes/pixel).
// Quadratic forms for 16 gaussians x 16 pixels per v_wmma_f32_16x16x4_f32.
// ---------------------------------------------------------------------------
__global__ __launch_bounds__(256) void gs_render(
    const v4f* __restrict__ ws4, const float* __restrict__ bg,
    const float* __restrict__ topleft, float* __restrict__ out) {
  __shared__ v4f s4[2304];  // 36 KB: geom | colal | inv11 (same layout as ws)
  const int tid = threadIdx.x;

  // ---- Stage all sorted gaussian params to LDS (CDNA5 async path if avail).
#if defined(__gfx1250__) && __has_builtin(__builtin_amdgcn_global_load_async_to_lds_b128)
  {
    typedef __attribute__((address_space(1))) v4i gv4i;  // prints as __device__
    typedef __attribute__((address_space(3))) v4i sv4i;  // prints as __shared__
    for (int i = tid; i < 2304; i += 256) {
      gv4i* src = (gv4i*)(unsigned long long)(ws4 + i);
      sv4i* dst = (sv4i*)(unsigned int)(unsigned long long)(s4 + i);
      __builtin_amdgcn_global_load_async_to_lds_b128(src, dst, 0, 0);
    }
#if __has_builtin(__builtin_amdgcn_s_wait_asynccnt)
    __builtin_amdgcn_s_wait_asynccnt(0);
#else
    asm volatile("s_wait_asynccnt 0" ::: "memory");
#endif
  }
#else
  for (int i = tid; i < 2304; i += 256) s4[i] = ws4[i];
#endif
  __syncthreads();

  const v4f*   sgeom  = s4;
  const v4f*   scolal = s4 + NG;
  const float* sinv11 = (const float*)(s4 + 2 * NG);

  const int wave = tid >> 5;
  const int lane = tid & 31;
  const int W    = blockIdx.x * 8 + wave;  // 0..4095 = 256 rows * 16 groups
  const int row  = W >> 4;
  const int xg   = W & 15;
  const int n    = lane & 15;              // pixel index within group == N col
  const bool low = lane < 16;
  const int pcol = xg * 16 + n;

  const float tl0 = topleft[0], tl1 = topleft[1];
  // Wave-centered coordinates: x0 = group center, y' == 0 (one row per wave).
  const float x0 = ((float)(xg * 16) + 8.0f) * (1.0f / 256.0f) - tl0;
  const float py = ((float)row + 0.5f) * (1.0f / 256.0f) - tl1;
  const float xr = ((float)n - 7.5f) * (1.0f / 256.0f);  // exact

  // B operand: per-pixel features [x'^2, x', 1, 0] over K, split across halves
  v2f bop;
  bop.x = low ? xr * xr : 1.0f;  // K0 | K2
  bop.y = low ? xr      : 0.0f;  // K1 | K3

  float T = 1.0f;
  float accr = 0.0f, accg = 0.0f, accb = 0.0f;

  for (int cb = 0; cb < NG; cb += 16) {
    // A operand: features of gaussian (cb+n): [inv00, gx, cc, 0] over K.
    v4f   gm  = sgeom[cb + n];
    float i11 = sinv11[cb + n];
    float mxr = gm.x - x0;
    float myr = gm.y - py;
    float i00 = gm.z, ioff = gm.w;
    float gx = -(2.0f * i00 * mxr + ioff * myr);
    float cc = (i00 * mxr) * mxr + (ioff * mxr) * myr + (i11 * myr) * myr;
    v2f aop;
    aop.x = low ? i00 : cc;    // K0 | K2
    aop.y = low ? gx  : 0.0f;  // K1 | K3

    // quad(m, p) for m = cb..cb+15 (rows), p = group pixels (cols).
    v8f q;
#if defined(__gfx1250__) && __has_builtin(__builtin_amdgcn_wmma_f32_16x16x4_f32)
    {
      v8f zero = {};
      q = __builtin_amdgcn_wmma_f32_16x16x4_f32(
          false, aop, false, bop, (short)0, zero, false, false);
    }
#else
    {
      const float px = xr + x0;
      #pragma unroll
      for (int v = 0; v < 8; ++v) {
        int gi = cb + v + (low ? 0 : 8);
        v4f g2 = sgeom[gi];
        float dx = px - g2.x, dy = py - g2.y;
        q[v] = g2.z * dx * dx + g2.w * dx * dy + sinv11[gi] * dy * dy;
      }
    }
#endif
    // Lane-local composite of 8 gaussians (low: m=0..7, high: m=8..15).
    float p = 1.0f, cr = 0.0f, cg = 0.0f, cbl = 0.0f;
    if (T > TTHR) {
      float qmin = q[0];
      #pragma unroll
      for (int v = 1; v < 8; ++v) qmin = fminf(qmin, q[v]);
      if (qmin < 36.0f) {  // exp(-18): (1-a) rounds to 1.0f => exact skip
        const int gsel = cb + (low ? 0 : 8);
        #pragma unroll
        for (int v = 0; v < 8; ++v) {
          v4f  cl  = scolal[gsel + v];
          float e  = __expf(-0.5f * q[v]);
          float ae = fminf(cl.x * e, ACLIP);
          float w  = p * ae;                 // local exclusive transmittance
          cr += w * cl.y; cg += w * cl.z; cbl += w * cl.w;
          p *= (1.0f - ae);
        }
      }
    }
    // Combine the two half-chunks; both lanes keep identical pixel state.
    float po  = __shfl_xor(p,   16, 32);
    float cro = __shfl_xor(cr,  16, 32);
    float cgo = __shfl_xor(cg,  16, 32);
    float cbo = __shfl_xor(cbl, 16, 32);
    float plo = low ? p   : po,  phi = low ? po  : p;
    float rlo = low ? cr  : cro, rhi = low ? cro : cr;
    float glo = low ? cg  : cgo, ghi = low ? cgo : cg;
    float blo = low ? cbl : cbo, bhi = low ? cbo : cbl;
    accr += T * (rlo + plo * rhi);
    accg += T * (glo + plo * ghi);
    accb += T * (blo + plo * bhi);
    T *= plo * phi;

    if (__all(T <= TTHR)) break;  // wave-uniform early exit
  }

  if (low) {
    const int o = (row * IMG + pcol) * 3;
    out[o + 0] = accr + T * bg[o + 0];
    out[o + 1] = accg + T * bg[o + 1];
    out[o + 2] = accb + T * bg[o + 2];
  }
}

// ---------------------------------------------------------------------------
extern "C" void kernel_launch(void* const* d_in, const int* in_sizes, int n_in,
                              void* d_out, int out_size, void* d_ws,
                              size_t ws_size, hipStream_t stream) {
  const float* mean    = (const float*)d_in[0];
  const float* cov     = (const float*)d_in[1];
  const float* color   = (const float*)d_in[2];
  const float* alpha   = (const float*)d_in[3];
  const float* depth   = (const float*)d_in[4];
  const float* bg      = (const float*)d_in[5];
  const float* topleft = (const float*)d_in[6];
  v4f*   ws4 = (v4f*)d_ws;
  float* out = (float*)d_out;

  gs_sort_prep<<<1, 1024, 0, stream>>>(mean, cov, color, alpha, depth, ws4);
  gs_render<<<512, 256, 0, stream>>>(ws4, bg, topleft, out);
}